// ConservationOfFeatureSimilarity_74251394613467
// MI455X (gfx1250) — compile-verified
//
#include <hip/hip_runtime.h>
#include <math.h>

#define BATCH   16
#define NTOK    196
#define DFZ     768
#define NFT     256
#define NPROTO  200
#define MTOT    (BATCH * NTOK)                 /* 3136 */
#define TROWS   (MTOT / 16)                    /* 196 tile rows */
#define CB32    (TROWS / 2)                    /* 98 column blocks of 32 */
#define NBLK32  9702                           /* sum_r (CB32 - r/2) */
#define TPB     8                              /* 16x32 blocks (waves) per workgroup */
#define NGRID   ((NBLK32 + TPB - 1) / TPB)     /* 1213 */
#define TOPK    5
#define CHUNKS  ((MTOT + 255) / 256)           /* 13 */
#define GAMMA_C 1.0f
#define EPS_C   1e-8f

typedef float v2f __attribute__((ext_vector_type(2)));
typedef float v8f __attribute__((ext_vector_type(8)));

// ---- register-resident sorted top-5 (descending value, ascending index on ties) ----
__device__ __forceinline__ void tk_push(float tv[TOPK], int tix[TOPK], float val, int idx) {
  float cv = val; int ci = idx;
#pragma unroll
  for (int q = 0; q < TOPK; ++q) {
    bool better = (cv > tv[q]) || ((cv == tv[q]) && (ci < tix[q]));
    float ov = tv[q]; int oi = tix[q];
    if (better) { tv[q] = cv; tix[q] = ci; cv = ov; ci = oi; }
  }
}

__device__ __forceinline__ void tk_wave_merge(float tv[TOPK], int tix[TOPK]) {
#pragma unroll
  for (int off = 1; off < 32; off <<= 1) {
    float ov[TOPK]; int oi[TOPK];
#pragma unroll
    for (int q = 0; q < TOPK; ++q) {
      ov[q] = __shfl_xor(tv[q], off, 32);
      oi[q] = __shfl_xor(tix[q], off, 32);
    }
#pragma unroll
    for (int q = 0; q < TOPK; ++q) tk_push(tv, tix, ov[q], oi[q]);
  }
}

// ---- row L2-normalization: one 256-thread block per row ----
__global__ __launch_bounds__(256) void normalize_rows(const float* __restrict__ src,
                                                      float* __restrict__ dst, int cols) {
  const int row = blockIdx.x;
  const float* s = src + (size_t)row * cols;
  float acc = 0.0f;
  for (int k = threadIdx.x; k < cols; k += 256) { float x = s[k]; acc += x * x; }
  __shared__ float red[256];
  red[threadIdx.x] = acc;
  __syncthreads();
  for (int st = 128; st > 0; st >>= 1) {
    if (threadIdx.x < st) red[threadIdx.x] += red[threadIdx.x + st];
    __syncthreads();
  }
  const float inv = 1.0f / fmaxf(sqrtf(red[0]), EPS_C);
  float* d = dst + (size_t)row * cols;
  for (int k = threadIdx.x; k < cols; k += 256) d[k] = s[k] * inv;
}

// ---- per-token prototype max + first-occurrence argmax ----
__global__ __launch_bounds__(256) void proto_max(const float* __restrict__ proto,
                                                 float* __restrict__ mps, int* __restrict__ mpi) {
  const int m = blockIdx.x * 256 + threadIdx.x;
  if (m >= MTOT) return;
  const int b = m / NTOK, n = m % NTOK;
  const float* base = proto + ((size_t)b * NPROTO) * NTOK + n;
  float best = base[0]; int bi = 0;
  for (int p = 1; p < NPROTO; ++p) {
    float v = base[(size_t)p * NTOK];
    if (v > best) { best = v; bi = p; }   // strict > keeps first occurrence
  }
  mps[m] = best; mpi[m] = bi;
}

// ---- fused dual-Gram (fp32 WMMA, 16x32 register-blocked) + ranking + top-5 ----
__global__ __launch_bounds__(256) void gram_rank_topk(
    const float* __restrict__ Xf, const float* __restrict__ Xz,
    const float* __restrict__ mps, const int* __restrict__ mpi,
    const int* __restrict__ labels,
    float* __restrict__ blkv, int* __restrict__ blki) {
  const int lane = threadIdx.x & 31;
  const int wave = threadIdx.x >> 5;
  const int lo = lane & 15;
  const int hi = lane >> 4;

  float tv[TOPK]; int tix[TOPK];
#pragma unroll
  for (int q = 0; q < TOPK; ++q) { tv[q] = -__builtin_inff(); tix[q] = 0x7fffffff; }

  const int t = blockIdx.x * TPB + wave;        // wave-uniform 16x32 block id
  if (t < NBLK32) {
    // decode row-major over rows r, column-blocks cb >= r/2 (row r has CB32 - r/2 blocks)
    int r = 0, rem = t;
    while (rem >= CB32 - (r >> 1)) { rem -= CB32 - (r >> 1); ++r; }
    const int cb = (r >> 1) + rem;
    const int i0 = r * 16;
    const int j0 = cb * 32;
    const int j1 = j0 + 16;

    // feature Gram 16x32 block (K = 256): A fragment shared by both j-tiles
    v8f cf0 = {}, cf1 = {};
    {
      const float* ar  = Xf + (size_t)(i0 + lo) * NFT + 2 * hi;
      const float* b0r = Xf + (size_t)(j0 + lo) * NFT + 2 * hi;
      const float* b1r = Xf + (size_t)(j1 + lo) * NFT + 2 * hi;
      for (int k = 0; k < NFT; k += 4) {
        v2f a  = *(const v2f*)(ar + k);
        v2f b0 = *(const v2f*)(b0r + k);
        v2f b1 = *(const v2f*)(b1r + k);
        cf0 = __builtin_amdgcn_wmma_f32_16x16x4_f32(false, a, false, b0, (short)0, cf0, false, false);
        cf1 = __builtin_amdgcn_wmma_f32_16x16x4_f32(false, a, false, b1, (short)0, cf1, false, false);
      }
    }
    // frozen Gram 16x32 block (K = 768)
    v8f cz0 = {}, cz1 = {};
    {
      const float* ar  = Xz + (size_t)(i0 + lo) * DFZ + 2 * hi;
      const float* b0r = Xz + (size_t)(j0 + lo) * DFZ + 2 * hi;
      const float* b1r = Xz + (size_t)(j1 + lo) * DFZ + 2 * hi;
      for (int k = 0; k < DFZ; k += 4) {
        v2f a  = *(const v2f*)(ar + k);
        v2f b0 = *(const v2f*)(b0r + k);
        v2f b1 = *(const v2f*)(b1r + k);
        cz0 = __builtin_amdgcn_wmma_f32_16x16x4_f32(false, a, false, b0, (short)0, cz0, false, false);
        cz1 = __builtin_amdgcn_wmma_f32_16x16x4_f32(false, a, false, b1, (short)0, cz1, false, false);
      }
    }

    // C/D layout: VGPR v, lanes0-15 -> M=v, lanes16-31 -> M=v+8, N = lane&15
    const int n0 = j0 + lo;
    const int n1 = j1 + lo;
    const int   mpin0 = mpi[n0],         mpin1 = mpi[n1];
    const int   labn0 = labels[n0 / NTOK], labn1 = labels[n1 / NTOK];
    const float mpsn0 = mps[n0],         mpsn1 = mps[n1];
#pragma unroll
    for (int v = 0; v < 8; ++v) {
      const int m = i0 + v + 8 * hi;
      const int   mpim = mpi[m];
      const int   labm = labels[m / NTOK];
      const float mpsm = mps[m];
      const bool val0 = (n0 > m) && (mpim == mpin0) && (labm != labn0);
      const bool val1 = (n1 > m) && (mpim == mpin1) && (labm != labn1);
      const float rk0 = val0 ? (cf0[v] - cz0[v]) * mpsm * mpsn0 : 0.0f;
      const float rk1 = val1 ? (cf1[v] - cz1[v]) * mpsm * mpsn1 : 0.0f;
      tk_push(tv, tix, rk0, m * MTOT + n0);
      tk_push(tv, tix, rk1, m * MTOT + n1);
    }
  }

  tk_wave_merge(tv, tix);

  __shared__ float sv[TPB * TOPK];
  __shared__ int   si[TPB * TOPK];
  if (lane == 0) {
#pragma unroll
    for (int q = 0; q < TOPK; ++q) { sv[wave * TOPK + q] = tv[q]; si[wave * TOPK + q] = tix[q]; }
  }
  __syncthreads();
  if (threadIdx.x == 0) {
    float bv[TOPK]; int bi[TOPK];
#pragma unroll
    for (int q = 0; q < TOPK; ++q) { bv[q] = -__builtin_inff(); bi[q] = 0x7fffffff; }
    for (int e = 0; e < TPB * TOPK; ++e) tk_push(bv, bi, sv[e], si[e]);
#pragma unroll
    for (int q = 0; q < TOPK; ++q) {
      blkv[blockIdx.x * TOPK + q] = bv[q];
      blki[blockIdx.x * TOPK + q] = bi[q];
    }
  }
}

// ---- merge block winners -> global top-5 -> 10 selected row indices ----
__global__ __launch_bounds__(256) void final_topk(const float* __restrict__ blkv,
                                                  const int* __restrict__ blki,
                                                  int* __restrict__ selrows) {
  float tv[TOPK]; int tix[TOPK];
#pragma unroll
  for (int q = 0; q < TOPK; ++q) { tv[q] = -__builtin_inff(); tix[q] = 0x7fffffff; }
  for (int e = threadIdx.x; e < NGRID * TOPK; e += 256) tk_push(tv, tix, blkv[e], blki[e]);
  tk_wave_merge(tv, tix);

  __shared__ float sv[8 * TOPK];
  __shared__ int   si[8 * TOPK];
  const int lane = threadIdx.x & 31, wave = threadIdx.x >> 5;
  if (lane == 0) {
#pragma unroll
    for (int q = 0; q < TOPK; ++q) { sv[wave * TOPK + q] = tv[q]; si[wave * TOPK + q] = tix[q]; }
  }
  __syncthreads();
  if (threadIdx.x == 0) {
    float bv[TOPK]; int bi[TOPK];
#pragma unroll
    for (int q = 0; q < TOPK; ++q) { bv[q] = -__builtin_inff(); bi[q] = 0x7fffffff; }
    for (int e = 0; e < 8 * TOPK; ++e) tk_push(bv, bi, sv[e], si[e]);
#pragma unroll
    for (int q = 0; q < TOPK; ++q) {
      selrows[2 * q]     = bi[q] / MTOT;   // row
      selrows[2 * q + 1] = bi[q] % MTOT;   // col
    }
  }
}

// ---- per selected row: sum_j |featdot(row,j) - frozdot(row,j)| (10 rows x 13 chunks) ----
__global__ __launch_bounds__(256) void row_diff(const float* __restrict__ Xf,
                                                const float* __restrict__ Xz,
                                                const int* __restrict__ selrows,
                                                float* __restrict__ partials) {
  __shared__ float srf[NFT];
  __shared__ float srz[DFZ];
  const int trow = blockIdx.y;          // 0..9
  const int chunk = blockIdx.x;         // 0..CHUNKS-1
  const int row = selrows[trow];
  for (int k = threadIdx.x; k < NFT; k += 256) srf[k] = Xf[(size_t)row * NFT + k];
  for (int k = threadIdx.x; k < DFZ; k += 256) srz[k] = Xz[(size_t)row * DFZ + k];
  __syncthreads();

  const int j = chunk * 256 + threadIdx.x;
  float val = 0.0f;
  if (j < MTOT) {
    float df = 0.0f, dz = 0.0f;
    const float* cf = Xf + (size_t)j * NFT;
    for (int k = 0; k < NFT; ++k) df = fmaf(srf[k], cf[k], df);
    const float* cz = Xz + (size_t)j * DFZ;
    for (int k = 0; k < DFZ; ++k) dz = fmaf(srz[k], cz[k], dz);
    val = fabsf(df - dz);
  }
  __shared__ float red[256];
  red[threadIdx.x] = val;
  __syncthreads();
  for (int st = 128; st > 0; st >>= 1) {
    if (threadIdx.x < st) red[threadIdx.x] += red[threadIdx.x + st];
    __syncthreads();
  }
  if (threadIdx.x == 0) partials[trow * CHUNKS + chunk] = red[0];
}

// ---- deterministic fixed-order final sum ----
__global__ void finish(const float* __restrict__ partials, float* __restrict__ out) {
  if (threadIdx.x == 0 && blockIdx.x == 0) {
    float s = 0.0f;
    for (int e = 0; e < 10 * CHUNKS; ++e) s += partials[e];
    out[0] = GAMMA_C * s / (float)(10 * MTOT);
  }
}

extern "C" void kernel_launch(void* const* d_in, const int* in_sizes, int n_in,
                              void* d_out, int out_size, void* d_ws, size_t ws_size,
                              hipStream_t stream) {
  const float* frozen = (const float*)d_in[0];   // (16,196,768)
  const float* feat   = (const float*)d_in[1];   // (16,196,256)
  const float* proto  = (const float*)d_in[2];   // (16,200,196)
  const int*   labels = (const int*)d_in[3];     // (16,)

  // workspace layout (all 4-byte types)
  float* Xf       = (float*)d_ws;                        // M*NFT
  float* Xz       = Xf + (size_t)MTOT * NFT;             // M*DFZ
  float* mps      = Xz + (size_t)MTOT * DFZ;             // M
  int*   mpi      = (int*)(mps + MTOT);                  // M
  float* blkv     = (float*)(mpi + MTOT);                // NGRID*TOPK
  int*   blki     = (int*)(blkv + (size_t)NGRID * TOPK); // NGRID*TOPK
  int*   selrows  = blki + (size_t)NGRID * TOPK;         // 10 (padded 16)
  float* partials = (float*)(selrows + 16);              // 10*CHUNKS

  normalize_rows<<<MTOT, 256, 0, stream>>>(feat, Xf, NFT);
  normalize_rows<<<MTOT, 256, 0, stream>>>(frozen, Xz, DFZ);
  proto_max<<<(MTOT + 255) / 256, 256, 0, stream>>>(proto, mps, mpi);
  gram_rank_topk<<<NGRID, 256, 0, stream>>>(Xf, Xz, mps, mpi, labels, blkv, blki);
  final_topk<<<1, 256, 0, stream>>>(blkv, blki, selrows);
  dim3 grd(CHUNKS, 10);
  row_diff<<<grd, 256, 0, stream>>>(Xf, Xz, selrows, partials);
  finish<<<1, 64, 0, stream>>>(partials, (float*)d_out);
}